// FNO2d_395136991934
// MI455X (gfx1250) — compile-verified
//
#include <hip/hip_runtime.h>
#include <math.h>

#define HH   256
#define WWp  256
#define HWP  65536
#define CC   64
#define NB   8
#define NM1  16
#define NM2  16
#define PI_F 3.14159265358979323846f

// Padded LDS row strides (floats) chosen so successive rows land on distinct banks:
//   conv wsh: 68  (68 mod 64 = 4  -> lanes 0..15 banks 4i..4i+1, half-wave +2 interleaves -> 64 distinct)
//   conv xs : 72  (2 rows = 144 mod 64 = 16 -> half-waves use disjoint 16-bank groups)
//   dftW xs : 260 (260 mod 64 = 4)
//   invW Fw : 264 (2 rows = 528 mod 64 = 16)
#define WS_S 68
#define XS_S 72
#define DW_S 260
#define FW_S 264

typedef float v2f __attribute__((ext_vector_type(2)));
typedef float v8f __attribute__((ext_vector_type(8)));

__device__ __forceinline__ float gelu_exact(float x) {
    return 0.5f * x * (1.0f + erff(x * 0.70710678118654752440f));
}

__device__ __forceinline__ v8f wmma_f32(v2f a, v2f b, v8f c) {
    // D = A(16x4 f32) * B(4x16 f32) + C(16x16 f32)
    return __builtin_amdgcn_wmma_f32_16x16x4_f32(false, a, false, b, (short)0, c, false, false);
}

// ---------------------------------------------------------------------------
// Twiddle / basis tables (computed once, L2-resident).
//   Ew  [w*32 + j]  : forward W-DFT basis. j<16 -> cos(2pi j w/256); j>=16 -> -sin(2pi (j-16) w/256)
//   Fw  [j*256 + w] : inverse W (irfft) basis. ky=j>>1; even j: s*cos, odd j: -s*sin (0 for ky==0 odd);
//                     s = (ky==0 ? 1 : 2)/256
//   cosH/sinH [h*32 + j] : H twiddles for kx = j<16 ? j : 224+j (i.e. 240..255)
// ---------------------------------------------------------------------------
__global__ void k_init_tables(float* __restrict__ Ew, float* __restrict__ Fw,
                              float* __restrict__ cosH, float* __restrict__ sinH) {
    int tid = blockIdx.x * blockDim.x + threadIdx.x;
    if (tid >= 256 * 32) return;
    int wv = tid >> 5;
    int j  = tid & 31;
    {
        int ky = (j < 16) ? j : (j - 16);
        int r  = (ky * wv) & 255;
        float th = (2.0f * PI_F / 256.0f) * (float)r;
        Ew[wv * 32 + j] = (j < 16) ? cosf(th) : -sinf(th);
    }
    {
        int ky = j >> 1;
        int r  = (ky * wv) & 255;
        float th = (2.0f * PI_F / 256.0f) * (float)r;
        float sc = ((ky == 0) ? 1.0f : 2.0f) * (1.0f / 256.0f);
        float v;
        if ((j & 1) == 0) v = sc * cosf(th);
        else              v = (ky == 0) ? 0.0f : -sc * sinf(th);
        Fw[j * 256 + wv] = v;
    }
    {
        int kx = (j < 16) ? j : (224 + j);
        int r  = (kx * wv) & 255;
        float th = (2.0f * PI_F / 256.0f) * (float)r;
        cosH[wv * 32 + j] = cosf(th);
        sinH[wv * 32 + j] = sinf(th);
    }
}

// ---------------------------------------------------------------------------
// Lifting: hdn[b,o,p] = p_b[o] + sum_l p_w[o,l]*x[b,l,p] + p_w[o,10]*gx + p_w[o,11]*gy
// ---------------------------------------------------------------------------
__global__ void k_lift(const float* __restrict__ x, const float* __restrict__ pw,
                       const float* __restrict__ pb, float* __restrict__ out) {
    int tid = blockIdx.x * blockDim.x + threadIdx.x;   // 524288
    int b = tid >> 16;
    int p = tid & (HWP - 1);
    float xin[12];
#pragma unroll
    for (int l = 0; l < 10; ++l) xin[l] = x[(size_t)(b * 10 + l) * HWP + p];
    xin[10] = (float)(p >> 8)  * (1.0f / 255.0f);
    xin[11] = (float)(p & 255) * (1.0f / 255.0f);
    for (int o = 0; o < CC; ++o) {
        float acc = pb[o];
#pragma unroll
        for (int l = 0; l < 12; ++l) acc += pw[o * 12 + l] * xin[l];
        out[(size_t)(b * CC + o) * HWP + p] = acc;
    }
}

// ---------------------------------------------------------------------------
// Instance-norm statistics per (b,c): mean and rstd over 65536 elems (float4 loads).
// ---------------------------------------------------------------------------
__global__ void k_stats(const float* __restrict__ in, float* __restrict__ mean,
                        float* __restrict__ rstd) {
    __shared__ float s1[256];
    __shared__ float s2[256];
    int bc = blockIdx.x;
    int t  = threadIdx.x;
    const float4* p = (const float4*)(in + (size_t)bc * HWP);
    float a = 0.0f, b = 0.0f;
    for (int i = t; i < HWP / 4; i += 256) {
        float4 v = p[i];
        a += v.x + v.y + v.z + v.w;
        b += v.x * v.x + v.y * v.y + v.z * v.z + v.w * v.w;
    }
    s1[t] = a; s2[t] = b;
    __syncthreads();
    for (int s = 128; s > 0; s >>= 1) {
        if (t < s) { s1[t] += s1[t + s]; s2[t] += s2[t + s]; }
        __syncthreads();
    }
    if (t == 0) {
        float m = s1[0] * (1.0f / HWP);
        float v = s2[0] * (1.0f / HWP) - m * m;
        mean[bc] = m;
        rstd[bc] = rsqrtf(v + 1e-5f);
    }
}

// ---------------------------------------------------------------------------
// Forward W-direction truncated DFT (WMMA GEMM):
//   T1[row=bc*256+h][j] = sum_w norm(x[b,c,h,w]) * Ew[w][j],  j: 0..15 re, 16..31 im
// ---------------------------------------------------------------------------
__global__ void k_dftW(const float* __restrict__ src, const float* __restrict__ mean,
                       const float* __restrict__ rstd, const float* __restrict__ Ew,
                       float* __restrict__ T1) {
    __shared__ __align__(16) float xs[32 * DW_S];   // ~32.5 KB
    int blk = blockIdx.x;            // 4096 = 512 bc * 8 segments
    int bc  = blk >> 3;
    int rbase = (blk & 7) * 32;
    int t = threadIdx.x;             // 128

    float m = mean[bc], r = rstd[bc];
    const float4* srcp = (const float4*)(src + (size_t)bc * HWP + (size_t)rbase * 256);
    for (int i = t; i < 32 * 64; i += 128) {      // 2048 float4
        int row = i >> 6, cv = (i & 63) * 4;
        float4 v = srcp[i];
        v.x = (v.x - m) * r; v.y = (v.y - m) * r; v.z = (v.z - m) * r; v.w = (v.w - m) * r;
        *(float4*)&xs[row * DW_S + cv] = v;
    }
    __syncthreads();

    int wid = t >> 5;
    int L   = t & 31;
    int mm16 = L & 15, hi = L >> 4, kb = hi * 2;
    int mt = wid >> 1;
    int nt = wid & 1;
    int r0 = mt * 16;

    v8f acc = {0.f, 0.f, 0.f, 0.f, 0.f, 0.f, 0.f, 0.f};
    for (int k0 = 0; k0 < 256; k0 += 4) {
        v2f a, b;
        a.x = xs[(r0 + mm16) * DW_S + k0 + kb];
        a.y = xs[(r0 + mm16) * DW_S + k0 + kb + 1];
        b.x = Ew[(k0 + kb) * 32 + nt * 16 + mm16];
        b.y = Ew[(k0 + kb + 1) * 32 + nt * 16 + mm16];
        acc = wmma_f32(a, b, acc);
    }
#pragma unroll
    for (int rr = 0; rr < 8; ++rr) {
        int mrow = rr + 8 * hi;
        int grow = bc * 256 + rbase + r0 + mrow;
        T1[(size_t)grow * 32 + nt * 16 + mm16] = acc[rr];
    }
}

// ---------------------------------------------------------------------------
// Forward H-direction DFT over 32 kept modes.
// ---------------------------------------------------------------------------
__global__ void k_dftH(const float* __restrict__ T1, const float* __restrict__ cosH,
                       const float* __restrict__ sinH, float* __restrict__ T2) {
    int tid = blockIdx.x * blockDim.x + threadIdx.x;   // 262144
    int ky = tid & 15;
    int kx = (tid >> 4) & 31;
    int bc = tid >> 9;
    float re = 0.0f, im = 0.0f;
    for (int h = 0; h < 256; ++h) {
        float a = T1[(size_t)(bc * 256 + h) * 32 + ky];
        float b = T1[(size_t)(bc * 256 + h) * 32 + 16 + ky];
        float c = cosH[h * 32 + kx];
        float s = sinH[h * 32 + kx];
        re += a * c + b * s;     // (a+ib)(c - i s)
        im += b * c - a * s;
    }
    int gi = (bc * 512 + kx * 16 + ky) * 2;
    T2[gi] = re; T2[gi + 1] = im;
}

// ---------------------------------------------------------------------------
// Per-mode complex channel mix: O[b,o,mode] = sum_i T2[b,i,mode] * W[i,o,mode]
// ---------------------------------------------------------------------------
__global__ void k_mix(const float* __restrict__ T2, const float* __restrict__ w1,
                      const float* __restrict__ w2, float* __restrict__ O) {
    __shared__ float wsm[CC * CC * 2];   // 32 KB
    __shared__ float ts[NB * CC * 2];    // 4 KB
    int mode = blockIdx.x;
    int kx = mode >> 4, ky = mode & 15;
    const float* wg = (kx < 16) ? w1 : w2;
    int xx = kx & 15;
    int t = threadIdx.x;

    for (int i = t; i < CC * CC; i += 256) {
        int ii = i >> 6, oo = i & 63;
        int gi = (((ii * CC + oo) * NM1 + xx) * NM2 + ky) * 2;
        wsm[i * 2]     = wg[gi];
        wsm[i * 2 + 1] = wg[gi + 1];
    }
    for (int i = t; i < NB * CC; i += 256) {
        int gi = (i * 512 + mode) * 2;
        ts[i * 2]     = T2[gi];
        ts[i * 2 + 1] = T2[gi + 1];
    }
    __syncthreads();

    for (int u = t; u < NB * CC; u += 256) {
        int b = u >> 6, o = u & 63;
        float re = 0.0f, im = 0.0f;
        for (int i = 0; i < CC; ++i) {
            float ar = ts[(b * CC + i) * 2], ai = ts[(b * CC + i) * 2 + 1];
            float wr = wsm[(i * CC + o) * 2], wi = wsm[(i * CC + o) * 2 + 1];
            re += ar * wr - ai * wi;
            im += ar * wi + ai * wr;
        }
        int gi = ((b * CC + o) * 512 + mode) * 2;
        O[gi] = re; O[gi + 1] = im;
    }
}

// ---------------------------------------------------------------------------
// Inverse H (complex ifft over 32 modes), scale 1/256.
// ---------------------------------------------------------------------------
__global__ void k_invH(const float* __restrict__ O, const float* __restrict__ cosH,
                       const float* __restrict__ sinH, float* __restrict__ G) {
    int tid = blockIdx.x * blockDim.x + threadIdx.x;   // 2097152
    int ky = tid & 15;
    int h  = (tid >> 4) & 255;
    int bo = tid >> 12;
    float re = 0.0f, im = 0.0f;
    for (int kx = 0; kx < 32; ++kx) {
        float orr = O[(size_t)(bo * 512 + kx * 16 + ky) * 2];
        float oi  = O[(size_t)(bo * 512 + kx * 16 + ky) * 2 + 1];
        float c = cosH[h * 32 + kx];
        float s = sinH[h * 32 + kx];
        re += orr * c - oi * s;    // (re+i im)(c + i s)
        im += orr * s + oi * c;
    }
    size_t gi = (size_t)(bo * 256 + h) * 32 + 2 * ky;
    G[gi]     = re * (1.0f / 256.0f);
    G[gi + 1] = im * (1.0f / 256.0f);
}

// ---------------------------------------------------------------------------
// Inverse W (irfft reconstruction) as WMMA GEMM: out(rows x 256) = G(rows x 32) * Fw(32 x 256)
// Fw staged in LDS (padded) to keep the K-loop off global memory.
// ---------------------------------------------------------------------------
__global__ void k_invW(const float* __restrict__ G, const float* __restrict__ Fw,
                       float* __restrict__ out) {
    __shared__ __align__(16) float fs[32 * FW_S];   // ~33 KB
    int blk = blockIdx.x;            // 1024
    int t = threadIdx.x;             // 256

    for (int i = t; i < 32 * 64; i += 256) {        // 2048 float4
        int row = i >> 6, cv = (i & 63) * 4;
        *(float4*)&fs[row * FW_S + cv] = ((const float4*)Fw)[i];
    }
    __syncthreads();

    int wid = t >> 5, L = t & 31;
    int m = L & 15, hi = L >> 4, kb = hi * 2;
    int rbase = blk * 128 + wid * 16;

    v2f a[8];
#pragma unroll
    for (int kk = 0; kk < 8; ++kk) {
        a[kk].x = G[(size_t)(rbase + m) * 32 + kk * 4 + kb];
        a[kk].y = G[(size_t)(rbase + m) * 32 + kk * 4 + kb + 1];
    }
    for (int nt = 0; nt < 16; ++nt) {
        v8f acc = {0.f, 0.f, 0.f, 0.f, 0.f, 0.f, 0.f, 0.f};
#pragma unroll
        for (int kk = 0; kk < 8; ++kk) {
            v2f b;
            b.x = fs[(kk * 4 + kb) * FW_S + nt * 16 + m];
            b.y = fs[(kk * 4 + kb + 1) * FW_S + nt * 16 + m];
            acc = wmma_f32(a[kk], b, acc);
        }
#pragma unroll
        for (int rr = 0; rr < 8; ++rr) {
            int mrow = rr + 8 * hi;
            out[(size_t)(rbase + mrow) * 256 + nt * 16 + m] = acc[rr];
        }
    }
}

// ---------------------------------------------------------------------------
// conv1x1 (64->64) with fused input instance-norm + bias + GELU.
// Tile: 64 out-ch x 64 pixels; WMMA f32 16x16x4, K=64; padded LDS.
// ---------------------------------------------------------------------------
__global__ void k_conv_ng(const float* __restrict__ src, const float* __restrict__ mean,
                          const float* __restrict__ rstd, const float* __restrict__ Wg,
                          const float* __restrict__ bias, float* __restrict__ dst) {
    __shared__ __align__(16) float xs[CC * XS_S];    // 18 KB
    __shared__ __align__(16) float wsh[CC * WS_S];   // 17 KB
    int blk = blockIdx.x;            // 8192
    int b = blk >> 10;
    int pbase = (blk & 1023) * 64;
    int t = threadIdx.x;

    if (t < CC && pbase + 64 < HWP)
        __builtin_prefetch(&src[(size_t)(b * CC + t) * HWP + pbase + 64], 0, 1);

    for (int i = t; i < CC * 16; i += 256) {         // 1024 float4 of W
        int o = i >> 4, cv = (i & 15) * 4;
        *(float4*)&wsh[o * WS_S + cv] = ((const float4*)Wg)[i];
    }
    for (int i = t; i < CC * 16; i += 256) {         // 1024 float4 of X tile
        int c = i >> 4, cv = (i & 15) * 4;
        int bc = b * CC + c;
        float4 v = *(const float4*)&src[(size_t)bc * HWP + pbase + cv];
        float mm = mean[bc], rr = rstd[bc];
        v.x = (v.x - mm) * rr; v.y = (v.y - mm) * rr; v.z = (v.z - mm) * rr; v.w = (v.w - mm) * rr;
        *(float4*)&xs[c * XS_S + cv] = v;
    }
    __syncthreads();

    int wid = t >> 5, L = t & 31;
    int m = L & 15, hi = L >> 4, kb = hi * 2;
    int obase = (wid & 3) * 16;
    int ptbase = (wid >> 2) * 2;

    v2f a[16];
#pragma unroll
    for (int kk = 0; kk < 16; ++kk) {
        a[kk].x = wsh[(obase + m) * WS_S + kk * 4 + kb];
        a[kk].y = wsh[(obase + m) * WS_S + kk * 4 + kb + 1];
    }
    for (int pi = 0; pi < 2; ++pi) {
        int pt = ptbase + pi;
        v8f acc = {0.f, 0.f, 0.f, 0.f, 0.f, 0.f, 0.f, 0.f};
#pragma unroll
        for (int kk = 0; kk < 16; ++kk) {
            v2f bb;
            bb.x = xs[(kk * 4 + kb) * XS_S + pt * 16 + m];
            bb.y = xs[(kk * 4 + kb + 1) * XS_S + pt * 16 + m];
            acc = wmma_f32(a[kk], bb, acc);
        }
#pragma unroll
        for (int rr = 0; rr < 8; ++rr) {
            int o = obase + rr + 8 * hi;
            float v = gelu_exact(acc[rr] + bias[o]);
            dst[(size_t)(b * CC + o) * HWP + pbase + pt * 16 + m] = v;
        }
    }
}

// ---------------------------------------------------------------------------
// Fused residual step: dst = [gelu]( W1@src1 + W0@src0 + b1 + b0 )
// Two K=64 phases sharing one xs buffer (LDS stays at ~52 KB).
// ---------------------------------------------------------------------------
__global__ void k_conv_dual(const float* __restrict__ src1, const float* __restrict__ src0,
                            const float* __restrict__ W1g, const float* __restrict__ W0g,
                            const float* __restrict__ b1g, const float* __restrict__ b0g,
                            int do_gelu, float* __restrict__ dst) {
    __shared__ __align__(16) float xs[CC * XS_S];     // 18 KB
    __shared__ __align__(16) float wshA[CC * WS_S];   // 17 KB
    __shared__ __align__(16) float wshB[CC * WS_S];   // 17 KB
    int blk = blockIdx.x;             // 8192
    int b = blk >> 10;
    int pbase = (blk & 1023) * 64;
    int t = threadIdx.x;

    if (t < CC && pbase + 64 < HWP)
        __builtin_prefetch(&src1[(size_t)(b * CC + t) * HWP + pbase + 64], 0, 1);

    for (int i = t; i < CC * 16; i += 256) {
        int o = i >> 4, cv = (i & 15) * 4;
        *(float4*)&wshA[o * WS_S + cv] = ((const float4*)W1g)[i];
        *(float4*)&wshB[o * WS_S + cv] = ((const float4*)W0g)[i];
    }

    int wid = t >> 5, L = t & 31;
    int m = L & 15, hi = L >> 4, kb = hi * 2;
    int obase = (wid & 3) * 16;
    int ptbase = (wid >> 2) * 2;

    v8f acc[2];
    acc[0] = (v8f){0.f, 0.f, 0.f, 0.f, 0.f, 0.f, 0.f, 0.f};
    acc[1] = (v8f){0.f, 0.f, 0.f, 0.f, 0.f, 0.f, 0.f, 0.f};

    for (int phase = 0; phase < 2; ++phase) {
        const float* sp = phase ? src0 : src1;
        const float* wl = phase ? wshB : wshA;
        __syncthreads();   // protect xs reuse (and cover initial weight staging)
        for (int i = t; i < CC * 16; i += 256) {
            int c = i >> 4, cv = (i & 15) * 4;
            float4 v = *(const float4*)&sp[(size_t)(b * CC + c) * HWP + pbase + cv];
            *(float4*)&xs[c * XS_S + cv] = v;
        }
        __syncthreads();

        v2f a[16];
#pragma unroll
        for (int kk = 0; kk < 16; ++kk) {
            a[kk].x = wl[(obase + m) * WS_S + kk * 4 + kb];
            a[kk].y = wl[(obase + m) * WS_S + kk * 4 + kb + 1];
        }
        for (int pi = 0; pi < 2; ++pi) {
            int pt = ptbase + pi;
            v8f acl = acc[pi];
#pragma unroll
            for (int kk = 0; kk < 16; ++kk) {
                v2f bb;
                bb.x = xs[(kk * 4 + kb) * XS_S + pt * 16 + m];
                bb.y = xs[(kk * 4 + kb + 1) * XS_S + pt * 16 + m];
                acl = wmma_f32(a[kk], bb, acl);
            }
            acc[pi] = acl;
        }
    }

    for (int pi = 0; pi < 2; ++pi) {
        int pt = ptbase + pi;
#pragma unroll
        for (int rr = 0; rr < 8; ++rr) {
            int o = obase + rr + 8 * hi;
            float v = acc[pi][rr] + b1g[o] + b0g[o];
            if (do_gelu) v = gelu_exact(v);
            dst[(size_t)(b * CC + o) * HWP + pbase + pt * 16 + m] = v;
        }
    }
}

// ---------------------------------------------------------------------------
// Projection head, fused per pixel: out = qb2 + sum_o qw2[o]*gelu(qb1[o] + qw1[o,:]·hdn[:,p])
// ---------------------------------------------------------------------------
__global__ void k_qhead(const float* __restrict__ hdn, const float* __restrict__ w1,
                        const float* __restrict__ b1, const float* __restrict__ w2,
                        const float* __restrict__ b2, float* __restrict__ out) {
    int tid = blockIdx.x * blockDim.x + threadIdx.x;   // 524288
    int b = tid >> 16;
    int p = tid & (HWP - 1);
    float xr[64];
#pragma unroll
    for (int c = 0; c < 64; ++c) xr[c] = hdn[(size_t)(b * CC + c) * HWP + p];
    float acc = b2[0];
    for (int o = 0; o < 256; ++o) {
        float s = b1[o];
#pragma unroll
        for (int k = 0; k < 64; ++k) s += w1[o * 64 + k] * xr[k];
        acc += w2[o] * gelu_exact(s);
    }
    out[tid] = acc;
}

// ---------------------------------------------------------------------------
extern "C" void kernel_launch(void* const* d_in, const int* in_sizes, int n_in,
                              void* d_out, int out_size, void* d_ws, size_t ws_size,
                              hipStream_t stream) {
    (void)in_sizes; (void)n_in; (void)out_size; (void)ws_size;

    const float* x    = (const float*)d_in[0];
    const float* p_w  = (const float*)d_in[1];
    const float* p_b  = (const float*)d_in[2];
    const float* q_w1 = (const float*)d_in[3];
    const float* q_b1 = (const float*)d_in[4];
    const float* q_w2 = (const float*)d_in[5];
    const float* q_b2 = (const float*)d_in[6];
    const float *sc_w1[4], *sc_w2[4], *mlp_w1[4], *mlp_b1[4], *mlp_w2[4], *mlp_b2[4], *w_w[4], *w_b[4];
    for (int i = 0; i < 4; ++i) {
        const int base = 7 + 8 * i;
        sc_w1[i]  = (const float*)d_in[base + 0];
        sc_w2[i]  = (const float*)d_in[base + 1];
        mlp_w1[i] = (const float*)d_in[base + 2];
        mlp_b1[i] = (const float*)d_in[base + 3];
        mlp_w2[i] = (const float*)d_in[base + 4];
        mlp_b2[i] = (const float*)d_in[base + 5];
        w_w[i]    = (const float*)d_in[base + 6];
        w_b[i]    = (const float*)d_in[base + 7];
    }

    float* ws = (float*)d_ws;
    size_t off = 0;
    const size_t FIELD = (size_t)NB * CC * HWP;     // 33,554,432 floats
    float* bufA = ws + off; off += FIELD;
    float* bufB = ws + off; off += FIELD;
    float* bufC = ws + off; off += FIELD;
    float* T1   = ws + off; off += (size_t)512 * 256 * 32;
    float* T2   = ws + off; off += (size_t)512 * 512 * 2;
    float* Obuf = ws + off; off += (size_t)512 * 512 * 2;
    float* G    = ws + off; off += (size_t)512 * 256 * 32;
    float* mean1 = ws + off; off += 512;
    float* rstd1 = ws + off; off += 512;
    float* mean2 = ws + off; off += 512;
    float* rstd2 = ws + off; off += 512;
    float* Ew   = ws + off; off += 256 * 32;
    float* Fw   = ws + off; off += 32 * 256;
    float* cosH = ws + off; off += 256 * 32;
    float* sinH = ws + off; off += 256 * 32;

    k_init_tables<<<32, 256, 0, stream>>>(Ew, Fw, cosH, sinH);
    k_lift<<<2048, 256, 0, stream>>>(x, p_w, p_b, bufA);

    float* cur = bufA;
    float* sA  = bufB;
    float* sB  = bufC;
    for (int i = 0; i < 4; ++i) {
        k_stats<<<512, 256, 0, stream>>>(cur, mean1, rstd1);
        k_dftW<<<4096, 128, 0, stream>>>(cur, mean1, rstd1, Ew, T1);
        k_dftH<<<1024, 256, 0, stream>>>(T1, cosH, sinH, T2);
        k_mix<<<512, 256, 0, stream>>>(T2, sc_w1[i], sc_w2[i], Obuf);
        k_invH<<<8192, 256, 0, stream>>>(Obuf, cosH, sinH, G);
        k_invW<<<1024, 256, 0, stream>>>(G, Fw, sA);
        k_stats<<<512, 256, 0, stream>>>(sA, mean2, rstd2);
        k_conv_ng<<<8192, 256, 0, stream>>>(sA, mean2, rstd2, mlp_w1[i], mlp_b1[i], sB);
        k_conv_dual<<<8192, 256, 0, stream>>>(sB, cur, mlp_w2[i], w_w[i], mlp_b2[i], w_b[i],
                                              (i < 3) ? 1 : 0, sA);
        float* tmp = cur; cur = sA; sA = tmp;
    }
    k_qhead<<<2048, 256, 0, stream>>>(cur, q_w1, q_b1, q_w2, q_b2, (float*)d_out);
}